// OperationEmbeddingLayer_71683004171058
// MI455X (gfx1250) — compile-verified
//
#include <hip/hip_runtime.h>

typedef float v2f __attribute__((ext_vector_type(2)));
typedef float v8f __attribute__((ext_vector_type(8)));

#define HID   128
#define LDX   17    // padded stride for 16-wide inputs (conflict-free, 17 coprime 64)
#define LDE   34    // padded stride for 32-wide concat buffer
#define LDH   132   // padded stride for 128-wide hidden activations
#define LDP   33    // padded stride for L3 partial buffer
#define MTILE 32    // nodes per block (2 WMMA M-tiles)

struct MlpW {
  const float* w1s; const float* b1;   // w*s = pre-swizzled fragment layouts
  const float* w2s; const float* b2;
  const float* w3s; const float* b3;   // zero-padded to 16 cols
};
struct AllW { MlpW p, s, r, m, c; };

__device__ __forceinline__ v8f wmma4(v2f a, v2f b, v8f c) {
  // D = A(16x4 f32) * B(4x16 f32) + C(16x16 f32)
  return __builtin_amdgcn_wmma_f32_16x16x4_f32(false, a, false, b, (short)0, c,
                                               false, false);
}

// branchless ELU: x>0 -> x ; x<=0 -> exp(x)-1   (v_exp_f32 path, no exec churn)
__device__ __forceinline__ float elu1(float x) {
  return fmaxf(x, 0.0f) + (__expf(fminf(x, 0.0f)) - 1.0f);
}

__global__ void zero_f32(float* __restrict__ p, long n) {
  long i = (long)blockIdx.x * blockDim.x + threadIdx.x;
  if (i < n) p[i] = 0.0f;
}

// ---- weight pre-swizzle: fragment pair (k, k+1) contiguous per lane ----
// dst[((kk*2 + lh)*128 + col)*2 + j] = src[(kk*4 + 2*lh + j)*128 + col]
struct Swz10 { const float* src[10]; int din[10]; long off[10]; };
__global__ void swz_hidden(Swz10 d, float* __restrict__ ws) {
  int m = blockIdx.y;
  int t = blockIdx.x * 256 + threadIdx.x;
  int elems = d.din[m] * HID;
  if (t >= elems) return;
  int j  = t & 1;
  int c  = (t >> 1) & 127;
  int lh = (t >> 8) & 1;
  int kk = t >> 9;
  int k  = kk * 4 + 2 * lh + j;
  ws[d.off[m] + t] = d.src[m][(long)k * HID + c];
}

// dst[((kk*4 + 2lh + b)*16 + col)*2 + j] (b = 0,1) = (col<8) ? src[(kk*4+2lh+j)*8 + col] : 0
struct Swz5 { const float* src[5]; long off[5]; };
__global__ void swz_w3(Swz5 d, float* __restrict__ ws) {
  int m = blockIdx.y;
  int t = blockIdx.x * 256 + threadIdx.x;   // 2048 elements per matrix
  if (t >= 2048) return;
  int j  = t & 1;
  int c  = (t >> 1) & 15;
  int lh = (t >> 6) & 1;
  int kk = t >> 7;
  int k  = kk * 4 + 2 * lh + j;
  ws[d.off[m] + t] = (c < 8) ? d.src[m][(long)k * 8 + c] : 0.0f;
}

// precedence edges: pred_sum[dst] += ops[src]; succ_sum[src] += ops[dst]; degrees
__global__ void prec_scatter(const int* __restrict__ pe, const float* __restrict__ ops,
                             float* __restrict__ pred_sum, float* __restrict__ succ_sum,
                             float* __restrict__ deg_in, float* __restrict__ deg_out,
                             int nE) {
  long t = (long)blockIdx.x * blockDim.x + threadIdx.x;
  int e = (int)(t >> 4);
  int c = (int)(t & 15);
  if (e >= nE) return;
  int s = pe[e];
  int d = pe[nE + e];
  atomicAdd(&pred_sum[(long)d * 16 + c], ops[(long)s * 16 + c]);
  atomicAdd(&succ_sum[(long)s * 16 + c], ops[(long)d * 16 + c]);
  if (c == 0) {
    atomicAdd(&deg_in[d], 1.0f);
    atomicAdd(&deg_out[s], 1.0f);
  }
}

// requirement edges: agg[op] += resources[res]
__global__ void req_scatter(const int* __restrict__ re, const float* __restrict__ res,
                            float* __restrict__ agg, int nE) {
  long t = (long)blockIdx.x * blockDim.x + threadIdx.x;
  int e = (int)(t >> 3);
  int c = (int)(t & 7);
  if (e >= nE) return;
  int op = re[e];
  int r  = re[nE + e];
  atomicAdd(&agg[(long)op * 8 + c], res[(long)r * 8 + c]);
}

__global__ __launch_bounds__(256) void gat_mlp(
    const float* __restrict__ ops,
    const float* __restrict__ pred_sum,
    const float* __restrict__ succ_sum,
    const float* __restrict__ agg,
    const float* __restrict__ deg_in,
    const float* __restrict__ deg_out,
    float* __restrict__ out,
    AllW wts, int N)
{
  __shared__ float sX[4][MTILE][LDX];   // p/s/r/m inputs (r zero-padded to 16)
  __shared__ float sE[MTILE][LDE];      // 32-wide concat for combining MLP
  __shared__ float sH1[MTILE][LDH];
  __shared__ float sH2[MTILE][LDH];
  __shared__ float sP[MTILE * LDP];     // L3 K-split partials

  const int tid  = threadIdx.x;
  const int lane = tid & 31;
  const int lh   = lane >> 4;                       // lane half (0/1)
  const int lm   = lane & 15;
  const int wave = tid >> 5;                        // 0..7 (uniform per wave)
  const int Nm2  = N - 2;
  const long nodeBase = 1 + (long)blockIdx.x * MTILE;

  // ---- stage 0: gather inputs into LDS ----
  for (int i = tid; i < MTILE * 16; i += 256) {
    int r = i >> 4, c = i & 15;
    long node = nodeBase + r;
    float xp = 0.f, xs = 0.f, xr = 0.f, xm = 0.f;
    if (node <= (long)Nm2) {
      float di = fmaxf(deg_in[node],  1.0f);
      float dd = fmaxf(deg_out[node], 1.0f);
      xp = pred_sum[node * 16 + c] / di;
      xs = succ_sum[node * 16 + c] / dd;
      xm = ops[node * 16 + c];
      xr = (c < 8) ? agg[node * 8 + c] : 0.0f;
    }
    sX[0][r][c] = xp; sX[1][r][c] = xs; sX[2][r][c] = xr; sX[3][r][c] = xm;
  }
  __syncthreads();

  const int col = wave * 16 + lm;     // this wave's hidden-column (L1/L2)
  const int mt  = wave & 1;           // L3: M-tile
  const int kq  = wave >> 1;          // L3: K-quarter (0..3)

#pragma unroll
  for (int mi = 0; mi < 5; ++mi) {
    const MlpW W = (mi == 0) ? wts.p : (mi == 1) ? wts.s : (mi == 2) ? wts.r
                 : (mi == 3) ? wts.m : wts.c;
    const float* Xin = (mi < 4) ? &sX[mi][0][0] : &sE[0][0];
    const int ldx = (mi < 4) ? LDX : LDE;
    const int din = (mi == 2) ? 8 : (mi == 4) ? 32 : 16;

    // ---- L1: X(32 x din) @ W1(din x 128) + b1, ELU -> sH1 ----
    {
      const float bias = W.b1[col];
      v8f acc0 = {bias, bias, bias, bias, bias, bias, bias, bias};
      v8f acc1 = acc0;
      const float* w1p = W.w1s + ((long)lh * HID + col) * 2;  // b64 per k-step
      const int ksteps = din >> 2;
      for (int kk = 0; kk < ksteps; ++kk) {
        int k = kk * 4 + 2 * lh;
        v2f b = *(const v2f*)(w1p + (long)kk * (HID * 4));
        v2f a0; a0.x = Xin[lm * ldx + k];        a0.y = Xin[lm * ldx + k + 1];
        v2f a1; a1.x = Xin[(16 + lm) * ldx + k]; a1.y = Xin[(16 + lm) * ldx + k + 1];
        acc0 = wmma4(a0, b, acc0);
        acc1 = wmma4(a1, b, acc1);
      }
#pragma unroll
      for (int v = 0; v < 8; ++v) {
        int row = v + 8 * lh;
        sH1[row][col]      = elu1(acc0[v]);
        sH1[16 + row][col] = elu1(acc1[v]);
      }
    }
    __syncthreads();

    // ---- L2: H1(32 x 128) @ W2(128 x 128) + b2, ELU -> sH2 ----
    {
      const float bias = W.b2[col];
      v8f acc0 = {bias, bias, bias, bias, bias, bias, bias, bias};
      v8f acc1 = acc0;
      const float* w2p = W.w2s + ((long)lh * HID + col) * 2;
      for (int kk = 0; kk < 32; ++kk) {
        int k = kk * 4 + 2 * lh;
        v2f b = *(const v2f*)(w2p + (long)kk * (HID * 4));
        v2f a0; a0.x = sH1[lm][k];      a0.y = sH1[lm][k + 1];
        v2f a1; a1.x = sH1[16 + lm][k]; a1.y = sH1[16 + lm][k + 1];
        acc0 = wmma4(a0, b, acc0);
        acc1 = wmma4(a1, b, acc1);
      }
#pragma unroll
      for (int v = 0; v < 8; ++v) {
        int row = v + 8 * lh;
        sH2[row][col]      = elu1(acc0[v]);
        sH2[16 + row][col] = elu1(acc1[v]);
      }
    }
    __syncthreads();

    // ---- L3: H2(32 x 128) @ W3(128 x 8pad16), K split over all 8 waves ----
    {
      v8f acc = {0.f, 0.f, 0.f, 0.f, 0.f, 0.f, 0.f, 0.f};
      const float* w3p = W.w3s + ((long)lh * 16 + lm) * 2;
      for (int kk = 0; kk < 8; ++kk) {
        int kt = kq * 8 + kk;
        int k  = kt * 4 + 2 * lh;
        v2f b = *(const v2f*)(w3p + (long)kt * 64);
        v2f a;  a.x = sH2[mt * 16 + lm][k]; a.y = sH2[mt * 16 + lm][k + 1];
        acc = wmma4(a, b, acc);
      }
      if (lm < 8) {
#pragma unroll
        for (int v = 0; v < 8; ++v) {
          int row = mt * 16 + v + 8 * lh;
          sP[row * LDP + kq * 8 + lm] = acc[v];
        }
      }
    }
    __syncthreads();

    // ---- reduce K-partials: one output element per thread ----
    {
      int row = tid >> 3;
      int c8  = tid & 7;
      float sum = sP[row * LDP + 0 * 8 + c8] + sP[row * LDP + 1 * 8 + c8]
                + sP[row * LDP + 2 * 8 + c8] + sP[row * LDP + 3 * 8 + c8]
                + W.b3[c8];
      if (mi < 4) {
        sE[row][mi * 8 + c8] = sum;
      } else {
        long node = nodeBase + row;
        if (node <= (long)Nm2) out[node * 8 + c8] = sum;
      }
    }
    __syncthreads();
  }
}

extern "C" void kernel_launch(void* const* d_in, const int* in_sizes, int n_in,
                              void* d_out, int out_size, void* d_ws, size_t ws_size,
                              hipStream_t stream) {
  const float* ops = (const float*)d_in[0];
  const float* res = (const float*)d_in[1];
  const int*   pe  = (const int*)d_in[2];
  const int*   re  = (const int*)d_in[3];
  const int N   = in_sizes[0] / 16;
  const int nPE = in_sizes[2] / 2;
  const int nRE = in_sizes[3] / 2;

  float* ws = (float*)d_ws;
  float* pred_sum = ws;                  // N*16
  float* succ_sum = ws + (long)N * 16;   // N*16
  float* agg      = ws + (long)N * 32;   // N*8
  float* deg_in   = ws + (long)N * 40;   // N
  float* deg_out  = ws + (long)N * 41;   // N
  const long wsN  = (long)N * 42;

  // swizzled-weight region layout
  const int dins[5] = {16, 16, 8, 16, 32};
  long o_w1[5], o_w2[5], o_w3[5];
  long off = wsN;
  for (int m = 0; m < 5; ++m) { o_w1[m] = off; off += (long)dins[m] * HID; }
  for (int m = 0; m < 5; ++m) { o_w2[m] = off; off += (long)HID * HID; }
  for (int m = 0; m < 5; ++m) { o_w3[m] = off; off += 2048; }

  // 1) zero accumulators + output (rows 0 and N-1 must stay zero)
  zero_f32<<<(unsigned)((wsN + 255) / 256), 256, 0, stream>>>(ws, wsN);
  zero_f32<<<(unsigned)(((long)out_size + 255) / 256), 256, 0, stream>>>(
      (float*)d_out, (long)out_size);

  // 2) swizzle weights into workspace (W1+W2 share the 128-col layout)
  Swz10 sh;
  for (int m = 0; m < 5; ++m) {
    sh.src[m]     = (const float*)d_in[4 + 6 * m + 0];  // *_w1
    sh.din[m]     = dins[m];
    sh.off[m]     = o_w1[m];
    sh.src[5 + m] = (const float*)d_in[4 + 6 * m + 2];  // *_w2
    sh.din[5 + m] = HID;
    sh.off[5 + m] = o_w2[m];
  }
  swz_hidden<<<dim3(64, 10), 256, 0, stream>>>(sh, ws);
  Swz5 s3;
  for (int m = 0; m < 5; ++m) {
    s3.src[m] = (const float*)d_in[4 + 6 * m + 4];      // *_w3
    s3.off[m] = o_w3[m];
  }
  swz_w3<<<dim3(8, 5), 256, 0, stream>>>(s3, ws);

  // 3) edge scatters (float atomics into L2)
  long pt = (long)nPE * 16;
  prec_scatter<<<(unsigned)((pt + 255) / 256), 256, 0, stream>>>(
      pe, ops, pred_sum, succ_sum, deg_in, deg_out, nPE);
  long rt = (long)nRE * 8;
  req_scatter<<<(unsigned)((rt + 255) / 256), 256, 0, stream>>>(re, res, agg, nRE);

  // 4) fused 5-MLP WMMA kernel over interior nodes
  auto mk = [&](int m) {
    MlpW w;
    w.w1s = ws + o_w1[m]; w.b1 = (const float*)d_in[4 + 6 * m + 1];
    w.w2s = ws + o_w2[m]; w.b2 = (const float*)d_in[4 + 6 * m + 3];
    w.w3s = ws + o_w3[m]; w.b3 = (const float*)d_in[4 + 6 * m + 5];
    return w;
  };
  AllW w;
  w.p = mk(0); w.s = mk(1); w.r = mk(2); w.m = mk(3); w.c = mk(4);

  int nInterior = N - 2;
  int nBlocks = (nInterior + MTILE - 1) / MTILE;
  gat_mlp<<<nBlocks, 256, 0, stream>>>(ops, pred_sum, succ_sum, agg,
                                       deg_in, deg_out, (float*)d_out, w, N);
}